// DCDLayer_35579509080779
// MI455X (gfx1250) — compile-verified
//
#include <hip/hip_runtime.h>
#include <hip/hip_bf16.h>

// ---------------------------------------------------------------------------
// Problem constants (match reference)
// ---------------------------------------------------------------------------
#define B_SEG 64
#define SEG_LEN 4096
#define NPTS (B_SEG * SEG_LEN)   // 262144
#define C_DIM 512
#define MID_DIM 2048
#define EPS 1e-5f

typedef __attribute__((ext_vector_type(2))) float v2f;
typedef __attribute__((ext_vector_type(4))) float v4f;
typedef __attribute__((ext_vector_type(8))) float v8f;

// ---------------------------------------------------------------------------
// K0: inclusive cumsum of npoint (B=64 values; trivial)
// ---------------------------------------------------------------------------
__global__ void dcd_cum_kernel(const int* __restrict__ npoint, int* __restrict__ cum) {
    if (threadIdx.x == 0 && blockIdx.x == 0) {
        int s = 0;
        for (int i = 0; i < B_SEG; ++i) { s += npoint[i]; cum[i] = s; }
    }
}

// ---------------------------------------------------------------------------
// K1: per-segment partial sums. grid = 64 segs * 4 chan-chunks * 8 row-chunks
//     = 2048 blocks of 128 threads. Channel-contiguous -> fully coalesced.
//     Deterministic (no atomics): partial[seg][rchunk][C].
// ---------------------------------------------------------------------------
__global__ void dcd_segsum_kernel(const float* __restrict__ x2,
                                  const int* __restrict__ cum,
                                  float* __restrict__ partial) {
    int blk    = blockIdx.x;
    int seg    = blk >> 5;        // 32 blocks per segment
    int sub    = blk & 31;
    int cchunk = sub >> 3;        // 0..3  (128 channels each)
    int rchunk = sub & 7;         // 0..7
    int start  = (seg == 0) ? 0 : cum[seg - 1];
    int end    = cum[seg];
    int len    = end - start;
    int cs     = (len + 7) >> 3;
    int r0     = start + rchunk * cs;
    int r1     = r0 + cs; if (r1 > end) r1 = end;
    int ch     = cchunk * 128 + threadIdx.x;

    float acc = 0.0f;
    for (int r = r0; r < r1; ++r)
        acc += x2[(size_t)r * C_DIM + ch];
    partial[(size_t)(seg * 8 + rchunk) * C_DIM + ch] = acc;
}

// ---------------------------------------------------------------------------
// K2: reduce partials -> mean_f [64,512]
// ---------------------------------------------------------------------------
__global__ void dcd_mean_kernel(const float* __restrict__ partial,
                                const int* __restrict__ npoint,
                                float* __restrict__ mean_f) {
    int gid = blockIdx.x * blockDim.x + threadIdx.x;   // 64*512 threads
    int seg = gid >> 9;
    int ch  = gid & (C_DIM - 1);
    float s = 0.0f;
    for (int r = 0; r < 8; ++r)
        s += partial[(size_t)(seg * 8 + r) * C_DIM + ch];
    mean_f[gid] = s / (float)npoint[seg];
}

// ---------------------------------------------------------------------------
// K3: GEMM1 (both branches): H1 = mean_f @ W1   M=64, K=512, N=2048
//     One wave per 16x16 output tile; V_WMMA_F32_16X16X4_F32 over K.
//     Tiles: 2 branches * 4 M-tiles * 128 N-tiles = 1024 waves = 128 blocks.
// ---------------------------------------------------------------------------
__global__ void dcd_gemm1_kernel(const float* __restrict__ mean_f,
                                 const float* __restrict__ W1a,
                                 const float* __restrict__ W1b,
                                 float* __restrict__ H1a,
                                 float* __restrict__ H1b) {
    int wave   = (blockIdx.x * blockDim.x + threadIdx.x) >> 5;
    int lane   = threadIdx.x & 31;
    int branch = wave >> 9;
    int rem    = wave & 511;
    int mt     = rem >> 7;        // 0..3
    int nt     = rem & 127;       // 0..127
    const float* W = branch ? W1b : W1a;
    float*       H = branch ? H1b : H1a;

    int m0 = mt * 16, n0 = nt * 16;
    int half = lane >> 4;         // 0: K pair {0,1}; 1: K pair {2,3}
    int l    = lane & 15;
    int kOff = half * 2;

    const float* arow = mean_f + (size_t)(m0 + l) * C_DIM;
    v8f acc = {};
    for (int k = 0; k < C_DIM; k += 4) {
        int k0 = k + kOff, k1 = k0 + 1;
        v2f a, b;
        a.x = arow[k0];
        a.y = arow[k1];
        b.x = W[(size_t)k0 * MID_DIM + n0 + l];
        b.y = W[(size_t)k1 * MID_DIM + n0 + l];
        acc = __builtin_amdgcn_wmma_f32_16x16x4_f32(
            /*neg_a=*/false, a, /*neg_b=*/false, b,
            /*c_mod=*/(short)0, acc, /*reuse_a=*/false, /*reuse_b=*/false);
    }
    // D layout: VGPR r -> M = r + 8*half, N = lane&15
    for (int r = 0; r < 8; ++r) {
        int m = m0 + r + half * 8;
        H[(size_t)m * MID_DIM + n0 + l] = acc[r];
    }
}

// ---------------------------------------------------------------------------
// K4: BN stats over the 64-row batch axis -> fold into scale/shift so the
//     BN-apply + ReLU fuses into GEMM2's A-operand load.
// ---------------------------------------------------------------------------
__global__ void dcd_bnstats_kernel(const float* __restrict__ H1a,
                                   const float* __restrict__ H1b,
                                   const float* __restrict__ g1a,
                                   const float* __restrict__ b1a,
                                   const float* __restrict__ g1b,
                                   const float* __restrict__ b1b,
                                   float* __restrict__ scale_a,
                                   float* __restrict__ shift_a,
                                   float* __restrict__ scale_b,
                                   float* __restrict__ shift_b) {
    int gid    = blockIdx.x * blockDim.x + threadIdx.x;  // 2 * 2048
    int branch = gid >> 11;
    int j      = gid & (MID_DIM - 1);
    const float* H = branch ? H1b : H1a;
    float s = 0.0f, s2 = 0.0f;
    for (int i = 0; i < B_SEG; ++i) {
        float v = H[(size_t)i * MID_DIM + j];
        s += v; s2 += v * v;
    }
    float m   = s * (1.0f / B_SEG);
    float var = s2 * (1.0f / B_SEG) - m * m;
    float g   = branch ? g1b[j] : g1a[j];
    float bb  = branch ? b1b[j] : b1a[j];
    float sc  = g * rsqrtf(var + EPS);
    float sh  = bb - m * sc;
    if (branch) { scale_b[j] = sc; shift_b[j] = sh; }
    else        { scale_a[j] = sc; shift_a[j] = sh; }
}

// ---------------------------------------------------------------------------
// K5: GEMM2 (both branches): out = act(relu(bn(H1)) @ W2)  M=64,K=2048,N=512
//     BN+ReLU fused into A-load. Branch a: ReLU; branch b: sigmoid(ReLU).
//     Tiles: 2 * 4 * 32 = 256 waves = 32 blocks.
// ---------------------------------------------------------------------------
__global__ void dcd_gemm2_kernel(const float* __restrict__ H1a,
                                 const float* __restrict__ H1b,
                                 const float* __restrict__ W2a,
                                 const float* __restrict__ W2b,
                                 const float* __restrict__ scale_a,
                                 const float* __restrict__ shift_a,
                                 const float* __restrict__ scale_b,
                                 const float* __restrict__ shift_b,
                                 float* __restrict__ out_mean,
                                 float* __restrict__ out_w) {
    int wave   = (blockIdx.x * blockDim.x + threadIdx.x) >> 5;
    int lane   = threadIdx.x & 31;
    int branch = wave >> 7;
    int rem    = wave & 127;
    int mt     = rem >> 5;        // 0..3
    int nt     = rem & 31;        // 0..31
    const float* H  = branch ? H1b : H1a;
    const float* W  = branch ? W2b : W2a;
    const float* sc = branch ? scale_b : scale_a;
    const float* sh = branch ? shift_b : shift_a;

    int m0 = mt * 16, n0 = nt * 16;
    int half = lane >> 4;
    int l    = lane & 15;
    int kOff = half * 2;

    const float* arow = H + (size_t)(m0 + l) * MID_DIM;
    v8f acc = {};
    for (int k = 0; k < MID_DIM; k += 4) {
        int k0 = k + kOff, k1 = k0 + 1;
        v2f a, b;
        float a0 = arow[k0] * sc[k0] + sh[k0];
        float a1 = arow[k1] * sc[k1] + sh[k1];
        a.x = fmaxf(a0, 0.0f);
        a.y = fmaxf(a1, 0.0f);
        b.x = W[(size_t)k0 * C_DIM + n0 + l];
        b.y = W[(size_t)k1 * C_DIM + n0 + l];
        acc = __builtin_amdgcn_wmma_f32_16x16x4_f32(
            false, a, false, b, (short)0, acc, false, false);
    }
    for (int r = 0; r < 8; ++r) {
        int m = m0 + r + half * 8;
        float v = fmaxf(acc[r], 0.0f);
        if (branch) {
            v = 1.0f / (1.0f + expf(-v));          // sigmoid(relu(x))
            out_w[(size_t)m * C_DIM + n0 + l] = v;
        } else {
            out_mean[(size_t)m * C_DIM + n0 + l] = v;
        }
    }
}

// ---------------------------------------------------------------------------
// K6: combine (bandwidth-bound pass): out = w[seg]*x*0.5 + 0.75*x + m[seg]
//     float4 (b128) streaming, non-temporal hints (write-once output; x2
//     (512MB) won't re-hit 192MB L2 anyway). seg via 6-step binary search.
// ---------------------------------------------------------------------------
__global__ void dcd_combine_kernel(const float* __restrict__ x2,
                                   const int* __restrict__ cum,
                                   const float* __restrict__ out_mean,
                                   const float* __restrict__ out_w,
                                   float* __restrict__ out) {
    size_t idx4 = (size_t)blockIdx.x * blockDim.x + threadIdx.x;  // one float4
    const size_t total4 = (size_t)NPTS * C_DIM / 4;
    if (idx4 >= total4) return;
    size_t row = idx4 / (C_DIM / 4);
    int    c4  = (int)(idx4 % (C_DIM / 4));

    // searchsorted(cum, row, side='right'): first i with cum[i] > row
    int lo = 0, hi = B_SEG - 1;
    while (lo < hi) {
        int mid = (lo + hi) >> 1;
        if ((long long)row < (long long)cum[mid]) hi = mid; else lo = mid + 1;
    }
    int seg = lo;

    v4f x = __builtin_nontemporal_load((const v4f*)x2 + idx4);
    v4f w = ((const v4f*)(out_w   + (size_t)seg * C_DIM))[c4];
    v4f m = ((const v4f*)(out_mean + (size_t)seg * C_DIM))[c4];
    v4f r;
    r.x = w.x * x.x * 0.5f + 0.75f * x.x + m.x;
    r.y = w.y * x.y * 0.5f + 0.75f * x.y + m.y;
    r.z = w.z * x.z * 0.5f + 0.75f * x.z + m.z;
    r.w = w.w * x.w * 0.5f + 0.75f * x.w + m.w;
    __builtin_nontemporal_store(r, (v4f*)out + idx4);
}

// ---------------------------------------------------------------------------
// Host launcher
// ---------------------------------------------------------------------------
extern "C" void kernel_launch(void* const* d_in, const int* in_sizes, int n_in,
                              void* d_out, int out_size, void* d_ws, size_t ws_size,
                              hipStream_t stream) {
    (void)in_sizes; (void)n_in; (void)out_size; (void)ws_size;
    const float* x2     = (const float*)d_in[0];
    const int*   npoint = (const int*)  d_in[1];
    const float* W1a    = (const float*)d_in[2];
    const float* g1a    = (const float*)d_in[3];
    const float* b1a    = (const float*)d_in[4];
    const float* W2a    = (const float*)d_in[5];
    const float* W1b    = (const float*)d_in[6];
    const float* g1b    = (const float*)d_in[7];
    const float* b1b    = (const float*)d_in[8];
    const float* W2b    = (const float*)d_in[9];
    float*       out    = (float*)d_out;

    // workspace layout (bytes)
    char* ws = (char*)d_ws;
    size_t off = 0;
    int*   cum      = (int*)(ws + off);  off += 256;                       // 64 ints, padded
    float* partial  = (float*)(ws + off); off += (size_t)B_SEG * 8 * C_DIM * 4;   // 1 MB
    float* mean_f   = (float*)(ws + off); off += (size_t)B_SEG * C_DIM * 4;       // 128 KB
    float* H1a      = (float*)(ws + off); off += (size_t)B_SEG * MID_DIM * 4;     // 512 KB
    float* H1b      = (float*)(ws + off); off += (size_t)B_SEG * MID_DIM * 4;     // 512 KB
    float* scale_a  = (float*)(ws + off); off += (size_t)MID_DIM * 4;
    float* shift_a  = (float*)(ws + off); off += (size_t)MID_DIM * 4;
    float* scale_b  = (float*)(ws + off); off += (size_t)MID_DIM * 4;
    float* shift_b  = (float*)(ws + off); off += (size_t)MID_DIM * 4;
    float* out_mean = (float*)(ws + off); off += (size_t)B_SEG * C_DIM * 4;
    float* out_w    = (float*)(ws + off); off += (size_t)B_SEG * C_DIM * 4;

    dcd_cum_kernel<<<1, 64, 0, stream>>>(npoint, cum);
    dcd_segsum_kernel<<<2048, 128, 0, stream>>>(x2, cum, partial);
    dcd_mean_kernel<<<(B_SEG * C_DIM) / 256, 256, 0, stream>>>(partial, npoint, mean_f);
    dcd_gemm1_kernel<<<128, 256, 0, stream>>>(mean_f, W1a, W1b, H1a, H1b);
    dcd_bnstats_kernel<<<(2 * MID_DIM) / 256, 256, 0, stream>>>(
        H1a, H1b, g1a, b1a, g1b, b1b, scale_a, shift_a, scale_b, shift_b);
    dcd_gemm2_kernel<<<32, 256, 0, stream>>>(
        H1a, H1b, W2a, W2b, scale_a, shift_a, scale_b, shift_b, out_mean, out_w);
    const size_t total4 = (size_t)NPTS * C_DIM / 4;
    dcd_combine_kernel<<<(unsigned)((total4 + 255) / 256), 256, 0, stream>>>(
        x2, cum, out_mean, out_w, out);
}